// RNN_78194174591059
// MI455X (gfx1250) — compile-verified
//
#include <hip/hip_runtime.h>
#include <cstddef>

typedef __attribute__((ext_vector_type(2))) float v2f;
typedef __attribute__((ext_vector_type(8))) float v8f;
typedef int v4i __attribute__((vector_size(16)));   // matches builtin prototype

#define B_     4
#define S_     4096
#define DMODEL 2048
#define NHEADS 32
#define HDIM   64
#define XGCOLS 4096          // X_SHAPE + G_SHAPE
#define STATE  2048
#define MROWS  (B_ * S_)     // 16384

// GEMM block tiling
#define BM  256
#define BN  64
#define BK  16
#define BKP 20               // padded LDS row stride (floats): float4-aligned,
                             // conflict-free for 16-lane b64 column reads

#if __has_builtin(__builtin_amdgcn_global_load_async_to_lds_b128)
#define HAVE_ASYNC_LDS 1
#else
#define HAVE_ASYNC_LDS 0
#endif

// async builtin param types (from hipcc diagnostic):
//   arg0: addrspace(1) v4i*   (global source)
//   arg1: addrspace(3) v4i*   (LDS destination)
//   arg2, arg3: int           (offset, cpol)
#define ASYNC_GPTR(p) ((__attribute__((address_space(1))) v4i*)(p))
#define ASYNC_LPTR(p) ((__attribute__((address_space(3))) v4i*)(p))

__device__ __forceinline__ void wait_asynccnt0() {
#if __has_builtin(__builtin_amdgcn_s_wait_asynccnt)
    __builtin_amdgcn_s_wait_asynccnt(0);
#else
    asm volatile("s_wait_asynccnt 0" ::: "memory");
#endif
}

__device__ __forceinline__ float silu_f(float v) {
    return v / (1.0f + __expf(-v));
}

// ---------------------------------------------------------------------------
// C[m,n] = sum_k A[m,k] * W[n,k]   (A: MxK row-major, W: NxK row-major)
// Block: 256 threads / 8 waves (4 M x 2 N), block tile 256x64, K-step 16.
// A/B tiles double-buffered in LDS (async-to-LDS when available); each wave
// computes a 64x32 tile via 4x2 accumulators of V_WMMA_F32_16X16X4_F32.
// Optional fused silu epilogue.
// ---------------------------------------------------------------------------
__global__ __launch_bounds__(256) void wmma_gemm_nt(
    const float* __restrict__ A, const float* __restrict__ W,
    float* __restrict__ C, int M, int N, int K, int do_silu)
{
    __shared__ float lA[2][BM * BKP];
    __shared__ float lB[2][BN * BKP];

    const int tid  = threadIdx.x;
    const int lane = tid & 31;
    const int wave = tid >> 5;
    const int wm   = wave >> 1;          // 0..3 (M)
    const int wn   = wave & 1;           // 0..1 (N)
    const int half = lane >> 4;          // 0/1 lane halves
    const int l15  = lane & 15;
    const int kk   = half * 2;

    const int bm0 = blockIdx.y * BM;
    const int bn0 = blockIdx.x * BN;

    // cooperative staging map: thread -> (row, 4-float column chunk)
    const int lrow = tid >> 2;           // 0..63
    const int lcol = (tid & 3) * 4;      // 0,4,8,12

    const float* Ald = A + (size_t)(bm0 + lrow) * K + lcol;
    const float* Wld = W + (size_t)(bn0 + lrow) * K + lcol;

    v8f acc[4][2];
    #pragma unroll
    for (int i = 0; i < 4; ++i)
        #pragma unroll
        for (int j = 0; j < 2; ++j)
            acc[i][j] = (v8f){0.f,0.f,0.f,0.f,0.f,0.f,0.f,0.f};

#if HAVE_ASYNC_LDS
    // -------- async global -> LDS staging (ASYNCcnt-tracked) --------
    auto stage_async = [&](int buf, int kt) {
        #pragma unroll
        for (int p = 0; p < 4; ++p)
            __builtin_amdgcn_global_load_async_to_lds_b128(
                ASYNC_GPTR(Ald + (size_t)(64 * p) * K + kt),
                ASYNC_LPTR(&lA[buf][(lrow + 64 * p) * BKP + lcol]),
                0, 0);
        __builtin_amdgcn_global_load_async_to_lds_b128(
            ASYNC_GPTR(Wld + kt),
            ASYNC_LPTR(&lB[buf][lrow * BKP + lcol]),
            0, 0);
    };
#else
    // -------- synchronous register-staged fallback --------
    float4 ra[4];
    float4 rb;
    auto fetch = [&](int kt) {
        #pragma unroll
        for (int p = 0; p < 4; ++p)
            ra[p] = *(const float4*)(Ald + (size_t)(64 * p) * K + kt);
        rb = *(const float4*)(Wld + kt);
    };
    auto store_stage = [&](int buf) {
        #pragma unroll
        for (int p = 0; p < 4; ++p)
            *(float4*)&lA[buf][(lrow + 64 * p) * BKP + lcol] = ra[p];
        *(float4*)&lB[buf][lrow * BKP + lcol] = rb;
    };
#endif

    auto compute_tile = [&](int buf) {
        const float* pa = &lA[buf][0];
        const float* pb = &lB[buf][0];
        #pragma unroll
        for (int ks = 0; ks < BK; ks += 4) {
            v2f a[4], b[2];
            #pragma unroll
            for (int i = 0; i < 4; ++i)
                a[i] = *(const v2f*)(pa + (wm * 64 + 16 * i + l15) * BKP + ks + kk);
            #pragma unroll
            for (int j = 0; j < 2; ++j)
                b[j] = *(const v2f*)(pb + (wn * 32 + 16 * j + l15) * BKP + ks + kk);
            #pragma unroll
            for (int i = 0; i < 4; ++i)
                #pragma unroll
                for (int j = 0; j < 2; ++j)
                    acc[i][j] = __builtin_amdgcn_wmma_f32_16x16x4_f32(
                        false, a[i], false, b[j], (short)0, acc[i][j],
                        false, false);
        }
    };

#if HAVE_ASYNC_LDS
    stage_async(0, 0);
    wait_asynccnt0();
    __syncthreads();
    int buf = 0;
    for (int kt = 0; kt < K; kt += BK) {
        if (kt + BK < K) stage_async(buf ^ 1, kt + BK);   // overlap with WMMAs
        compute_tile(buf);
        if (kt + BK < K) wait_asynccnt0();
        __syncthreads();
        buf ^= 1;
    }
#else
    fetch(0);
    store_stage(0);
    __syncthreads();
    int buf = 0;
    for (int kt = 0; kt < K; kt += BK) {
        if (kt + BK < K) fetch(kt + BK);                  // loads fly over WMMAs
        compute_tile(buf);
        if (kt + BK < K) store_stage(buf ^ 1);
        __syncthreads();
        buf ^= 1;
    }
#endif

    // C/D layout: VGPR r -> row (bm + 16*i + half*8 + r), col (bn + 16*j + l15)
    #pragma unroll
    for (int i = 0; i < 4; ++i) {
        #pragma unroll
        for (int j = 0; j < 2; ++j) {
            float* crow = C + (size_t)(bm0 + wm * 64 + 16 * i + half * 8) * N
                            + (bn0 + wn * 32 + 16 * j + l15);
            #pragma unroll
            for (int r = 0; r < 8; ++r) {
                float v = acc[i][j][r];
                if (do_silu) v = silu_f(v);
                crow[(size_t)r * N] = v;
            }
        }
    }
}

// ---------------------------------------------------------------------------
// Recurrent scan: h = tanh(h @ W_state[n] + silu(xh_t)) per (batch, head).
// One 64-thread block per (b, n); lane e owns h[e] and a register-resident
// column W_state[n, :, e]; h broadcast via LDS with split barriers per step.
// ---------------------------------------------------------------------------
__global__ __launch_bounds__(64) void scan_kernel(
    const float* __restrict__ xp,     // [MROWS, 4096] (cols 0..2047 = silu(xh))
    const float* __restrict__ Wst,    // [32, 64, 64]
    float* __restrict__ y)            // [MROWS, 2048]
{
    const int bn = blockIdx.x;        // 0..127
    const int b  = bn >> 5;
    const int n  = bn & 31;
    const int e  = threadIdx.x;       // 0..63

    float w[HDIM];
    #pragma unroll
    for (int d = 0; d < HDIM; ++d)
        w[d] = Wst[(size_t)n * HDIM * HDIM + d * HDIM + e];

    __shared__ float h[HDIM];
    h[e] = 0.0f;
    __syncthreads();

    const float* xcol = xp + (size_t)b * S_ * XGCOLS + n * HDIM + e;
    float*       ycol = y  + (size_t)b * S_ * STATE  + n * HDIM + e;

    for (int t = 0; t < S_; ++t) {
        float acc = xcol[(size_t)t * XGCOLS];
        #pragma unroll
        for (int d = 0; d < HDIM; ++d)
            acc += h[d] * w[d];
        float hn = tanhf(acc);
        __syncthreads();              // all reads of h done
        h[e] = hn;
        ycol[(size_t)t * STATE] = hn;
        __syncthreads();              // writes visible before next dot
    }
}

// ---------------------------------------------------------------------------
// y = y * silu(g); RMSNorm over 2048 with norm_w. One block per row.
// ---------------------------------------------------------------------------
__global__ __launch_bounds__(256) void norm_kernel(
    const float* __restrict__ yin,    // [MROWS, 2048]
    const float* __restrict__ gs,     // [MROWS, 4096], gate half pre-silu'd
    const float* __restrict__ nw,     // [2048]
    float* __restrict__ yout)         // [MROWS, 2048] (may alias yin)
{
    const int row  = blockIdx.x;
    const int tid  = threadIdx.x;
    const int lane = tid & 31;
    const int wv   = tid >> 5;

    const float* yr = yin + (size_t)row * STATE;
    const float* gr = gs  + (size_t)row * XGCOLS + STATE;

    float v[8];
    float ss = 0.0f;
    #pragma unroll
    for (int i = 0; i < 8; ++i) {
        int c = tid + i * 256;
        float t = yr[c] * gr[c];
        v[i] = t;
        ss += t * t;
    }

    #pragma unroll
    for (int off = 16; off > 0; off >>= 1)
        ss += __shfl_xor(ss, off, 32);

    __shared__ float red[8];
    if (lane == 0) red[wv] = ss;
    __syncthreads();
    if (tid == 0) {
        float t = 0.0f;
        #pragma unroll
        for (int i = 0; i < 8; ++i) t += red[i];
        red[0] = rsqrtf(t * (1.0f / (float)STATE) + 1e-6f);
    }
    __syncthreads();
    const float r = red[0];

    #pragma unroll
    for (int i = 0; i < 8; ++i) {
        int c = tid + i * 256;
        yout[(size_t)row * STATE + c] = v[i] * r * nw[c];
    }
}

// ---------------------------------------------------------------------------
extern "C" void kernel_launch(void* const* d_in, const int* in_sizes, int n_in,
                              void* d_out, int out_size, void* d_ws, size_t ws_size,
                              hipStream_t stream)
{
    const float* x       = (const float*)d_in[0];  // [4,4096,2048]
    const float* W_in    = (const float*)d_in[1];  // [4096,2048]
    const float* W_state = (const float*)d_in[2];  // [32,64,64]
    const float* norm_w  = (const float*)d_in[3];  // [2048]
    const float* W_out   = (const float*)d_in[4];  // [2048,2048]
    float* out = (float*)d_out;                    // [4,4096,2048]

    float* xp = (float*)d_ws;                      // [16384, 4096]
    float* y  = xp + (size_t)MROWS * XGCOLS;       // [16384, 2048]

    dim3 blk(256);

    // xp = silu(x @ W_in^T)   M=16384 N=4096 K=2048
    wmma_gemm_nt<<<dim3(XGCOLS / BN, MROWS / BM), blk, 0, stream>>>(
        x, W_in, xp, MROWS, XGCOLS, DMODEL, 1);

    // sequential recurrence over S, parallel over B*heads
    scan_kernel<<<dim3(B_ * NHEADS), dim3(64), 0, stream>>>(xp, W_state, y);

    // gate + RMSNorm (in place on y)
    norm_kernel<<<dim3(MROWS), blk, 0, stream>>>(y, xp, norm_w, y);

    // out = y @ W_out^T       M=16384 N=2048 K=2048
    wmma_gemm_nt<<<dim3(STATE / BN, MROWS / BM), blk, 0, stream>>>(
        y, W_out, out, MROWS, STATE, STATE, 0);
}